// CRFLayer_72327249265017
// MI455X (gfx1250) — compile-verified
//
#include <hip/hip_runtime.h>
#include <hip/hip_bf16.h>

// CRF forward (partition function) for MI455X / gfx1250.
// S=512 sequential steps; per step: (1024x64) exp-alpha  @  exp(T)^T (64x64)
// done as f32 WMMA 16x16x4, feats streamed by the Tensor Data Mover into
// double-buffered LDS tiles, then log + feat + masked select.

typedef __attribute__((ext_vector_type(2))) float v2f;
typedef __attribute__((ext_vector_type(8))) float v8f;
typedef __attribute__((ext_vector_type(4))) unsigned int u32x4;
typedef __attribute__((ext_vector_type(8))) int i32x8;
typedef __attribute__((ext_vector_type(4))) int i32x4;

#define SEQ_LEN 512
#define BATCH   1024
#define TAG     64
#define NEG_INF (-10000.0f)

// Workgroup: 4 waves (128 threads). Each wave owns 16 batch rows.
// Grid: 1024 / 64 = 16 workgroups -> one wave per SIMD32 on 16 WGPs.

// Issue a TDM load of a 16x64 f32 tile (16 batch rows x 64 tags, row stride
// 64 floats) from global feats into LDS at byte offset lds_off.
// D# layout per CDNA5 ISA.
// This toolchain exposes the 6-arg builtin form:
//   (u32x4 g0, i32x8 g1, i32x4 g2, i32x4 g3, i32x8 pad, i32 cpol)
__device__ __forceinline__ void tdm_load_tile(const float* gptr, unsigned lds_off) {
    unsigned lo = (unsigned)__builtin_amdgcn_readfirstlane((int)(unsigned)(uintptr_t)gptr);
    unsigned hi = (unsigned)__builtin_amdgcn_readfirstlane((int)(unsigned)((uintptr_t)gptr >> 32));
    unsigned la = (unsigned)__builtin_amdgcn_readfirstlane((int)lds_off);

    // group0: count=1 | lds_addr | global_addr[56:0] | type=2 ("image")
    u32x4 g0 = { 1u, la, lo, ((hi & 0x01FFFFFFu) | (2u << 30)) };
    // group1: data_size=4B; tensor_dim0=64; tensor_dim1=1024; tile_dim0=64;
    //         tile_dim1=16 rows; tensor_dim0_stride=64
    i32x8 g1 = { (int)(2u << 16),
                 (int)(64u << 16),
                 (int)(1024u << 16),
                 (int)(64u << 16),
                 16, 64, 0, 0 };
    // groups 2/3: benign 1-extent higher dims (tile_dim2/3/4 = 0 -> unused)
    i32x4 g2 = { 1, 1, 0, 0 };
    i32x4 g3 = { 0, (int)(1u << 16), 0, 0 };
    i32x8 gz = { 0, 0, 0, 0, 0, 0, 0, 0 };
    __builtin_amdgcn_tensor_load_to_lds(g0, g1, g2, g3, gz, 0);
}

__global__ __launch_bounds__(128) void crf_fwd_kernel(
    const float* __restrict__ feats,   // [S, B, T]
    const float* __restrict__ mask,    // [S, B]
    const float* __restrict__ trans,   // [T, T]
    float* __restrict__ out)           // [B]
{
    // Packed B-matrix exp(T)^T in LDS:
    // B[k][n] = exp(trans[n*64 + k]); packed so (k, k+1) pairs are adjacent:
    // float index = (k>>1)*128 + n*2 + (k&1)  -> one ds_load_b64 per fragment.
    __shared__ float sBT[TAG * TAG];           // 16 KB
    __shared__ float sEA[4][16 * TAG];         // per-wave exp(alpha - m), 4x4 KB
    __shared__ float sFeat[4][2][16 * TAG];    // per-wave double-buffered feat tiles, 4x8 KB

    const int tid  = threadIdx.x;
    const int wave = tid >> 5;
    const int lane = tid & 31;
    const int half = lane >> 4;    // 0: lanes 0-15, 1: lanes 16-31
    const int l    = lane & 15;
    const int baseb = blockIdx.x * 64 + wave * 16;   // first batch row of this wave

    // ---- build exp(transition)^T (packed) cooperatively ----
    for (int idx = tid; idx < TAG * TAG; idx += blockDim.x) {
        int n = idx >> 6;          // output tag j
        int k = idx & 63;          // input tag i
        sBT[(k >> 1) * 128 + n * 2 + (k & 1)] = __expf(trans[n * TAG + k]);
    }
    __syncthreads();

    // ---- kick off TDM double-buffer: feat tiles for t=0 and t=1 ----
    const unsigned fb_off0 = (unsigned)(uintptr_t)&sFeat[wave][0][0];
    const unsigned fb_off1 = (unsigned)(uintptr_t)&sFeat[wave][1][0];
    tdm_load_tile(&feats[(size_t)baseb * TAG], fb_off0);
    tdm_load_tile(&feats[((size_t)BATCH + baseb) * TAG], fb_off1);

    // ---- hoist all B fragments into registers (constant over all 512 steps) ----
    // Guessed B layout (mirrors documented 32-bit A 16x4 layout):
    //   VGPR0: lanes 0-15 hold K=0, lanes 16-31 hold K=2; VGPR1: K=1 / K=3.
    v2f Bf[16][4];
#pragma unroll
    for (int k = 0; k < 16; ++k)
#pragma unroll
        for (int nt = 0; nt < 4; ++nt)
            Bf[k][nt] = ((const v2f*)sBT)[(2 * k + half) * 64 + nt * 16 + l];

    // ---- alpha in WMMA C-layout registers ----
    // acc[nt][v]: batch row = v + 8*half, tag = nt*16 + l
    v8f acc[4];
#pragma unroll
    for (int nt = 0; nt < 4; ++nt)
#pragma unroll
        for (int v = 0; v < 8; ++v)
            acc[nt][v] = (nt == 0 && l == 0) ? 0.0f : NEG_INF;  // START tag = 0

    float* myEA = &sEA[wave][0];

    for (int t = 0; t < SEQ_LEN; ++t) {
        __builtin_prefetch(&mask[(size_t)(t + 1) * BATCH + baseb], 0, 1);

        // 1) per-row max of alpha (across 4 tiles, then across the 16 lanes of each half)
        float rm[8];
#pragma unroll
        for (int v = 0; v < 8; ++v) {
            float r = fmaxf(fmaxf(acc[0][v], acc[1][v]), fmaxf(acc[2][v], acc[3][v]));
            r = fmaxf(r, __shfl_xor(r, 1, 32));
            r = fmaxf(r, __shfl_xor(r, 2, 32));
            r = fmaxf(r, __shfl_xor(r, 4, 32));
            r = fmaxf(r, __shfl_xor(r, 8, 32));
            rm[v] = r;
        }

        // 2) stage exp(alpha - m) into LDS, [row][tag] row-major (row = v + 8*half)
#pragma unroll
        for (int v = 0; v < 8; ++v) {
            const int row = v + 8 * half;
#pragma unroll
            for (int nt = 0; nt < 4; ++nt)
                myEA[row * TAG + nt * 16 + l] = __expf(acc[nt][v] - rm[v]);
        }
        // wave-local LDS RAW: wait for our stores before reading A fragments
        asm volatile("s_wait_dscnt 0" ::: "memory");

        // 3) P = exp(alpha-m) @ exp(T)^T via 64x V_WMMA_F32_16X16X4_F32
        v8f p[4];
#pragma unroll
        for (int nt = 0; nt < 4; ++nt)
#pragma unroll
            for (int v = 0; v < 8; ++v)
                p[nt][v] = 0.0f;

#pragma unroll
        for (int k = 0; k < 16; ++k) {
            // A fragment: row = l, cols (4k + 2*half, 4k + 2*half + 1) -> float2
            v2f a2 = ((const v2f*)myEA)[l * 32 + 2 * k + half];
#pragma unroll
            for (int nt = 0; nt < 4; ++nt)
                p[nt] = __builtin_amdgcn_wmma_f32_16x16x4_f32(
                    /*neg_a=*/false, a2, /*neg_b=*/false, Bf[k][nt],
                    /*c_mod=*/(short)0, p[nt], /*reuse_a=*/false, /*reuse_b=*/false);
        }

        // ensure this step's TDM feat tile has landed in LDS
        if (t + 1 < SEQ_LEN) __builtin_amdgcn_s_wait_tensorcnt(1);
        else                 __builtin_amdgcn_s_wait_tensorcnt(0);

        // 4) new_alpha = feat + m + log(P); masked select against old alpha
        const float* fb = &sFeat[wave][t & 1][0];
#pragma unroll
        for (int v = 0; v < 8; ++v) {
            const int gb = baseb + v + 8 * half;
            const float mk = mask[(size_t)t * BATCH + gb];
#pragma unroll
            for (int nt = 0; nt < 4; ++nt) {
                const float f = fb[(v + 8 * half) * TAG + nt * 16 + l];
                float s = fmaxf(p[nt][v], 1e-30f);   // START row sums to 0 -> clamp
                const float na = __logf(s) + rm[v] + f;
                acc[nt][v] = (mk > 0.5f) ? na : acc[nt][v];
            }
        }

        // 5) re-issue TDM for step t+2 into the buffer we just consumed
        //    (WAR: make sure our LDS reads of this buffer are complete first)
        if (t + 2 < SEQ_LEN) {
            asm volatile("s_wait_dscnt 0" ::: "memory");
            tdm_load_tile(&feats[((size_t)(t + 2) * BATCH + baseb) * TAG],
                          (t & 1) ? fb_off1 : fb_off0);
        }
    }

    // ---- epilogue: out[b] = logsumexp_j(alpha[b,j] + trans[END=1, j]) ----
#pragma unroll
    for (int v = 0; v < 8; ++v) {
        float vals[4];
        float r = -3.4e38f;
#pragma unroll
        for (int nt = 0; nt < 4; ++nt) {
            vals[nt] = acc[nt][v] + trans[1 * TAG + nt * 16 + l];
            r = fmaxf(r, vals[nt]);
        }
        r = fmaxf(r, __shfl_xor(r, 1, 32));
        r = fmaxf(r, __shfl_xor(r, 2, 32));
        r = fmaxf(r, __shfl_xor(r, 4, 32));
        r = fmaxf(r, __shfl_xor(r, 8, 32));
        float s = 0.0f;
#pragma unroll
        for (int nt = 0; nt < 4; ++nt)
            s += __expf(vals[nt] - r);
        s += __shfl_xor(s, 1, 32);
        s += __shfl_xor(s, 2, 32);
        s += __shfl_xor(s, 4, 32);
        s += __shfl_xor(s, 8, 32);
        if (l == 0)
            out[baseb + v + 8 * half] = __logf(s) + r;
    }
}

extern "C" void kernel_launch(void* const* d_in, const int* in_sizes, int n_in,
                              void* d_out, int out_size, void* d_ws, size_t ws_size,
                              hipStream_t stream) {
    (void)in_sizes; (void)n_in; (void)d_ws; (void)ws_size; (void)out_size;
    const float* feats = (const float*)d_in[0];
    const float* mask  = (const float*)d_in[1];
    const float* trans = (const float*)d_in[2];
    float* out = (float*)d_out;

    dim3 grid(BATCH / 64);   // 16 workgroups
    dim3 block(128);         // 4 waves, 16 batch rows each
    crf_fwd_kernel<<<grid, block, 0, stream>>>(feats, mask, trans, out);
}